// SparseMMLayer_53257594470705
// MI455X (gfx1250) — compile-verified
//
#include <hip/hip_runtime.h>

// ELL sparse matmul: C[b,n,d] = sum_k A[b,n,k] * B[b, idx[b,n,k], d]
// BATCH=4, N=50000, K=32, D=64, fp32. Gather-bound: 1.64 GB of L2-resident
// row gathers vs 0.82 GFLOP (~0.5 FLOP/B). One wave32 per output row n;
// accumulation on the matrix unit via V_WMMA_F32_16X16X4_F32 (f32 matches
// reference precision; XDL has ~1000x headroom so idle M-rows cost nothing).
//
// d-tile t covers interleaved columns {d : d % 4 == t}: lane j of all four
// B-fragments needs columns 4j..4j+3 of a gathered row -> one
// global_load_b128 per gathered row-pair (512B/wave instruction, lands in
// fragment layout), one nontemporal global_store_b128 for the output.
//
// Batch id derives from blockIdx.x only (6250 blocks/batch, exact) so the
// B-slice base is uniform (SGPR pair). Gather offsets are computed as 32-bit
// BYTE offsets off a char* base (provably < 12.8MB), giving the backend the
// exact base + zext(i32) pattern for saddr-form global_load_b128 -> one
// v_lshl_or_b32 per gather, zero 64-bit vector address math.
// All 32 lane-broadcasts (ds_bpermute) are hoisted up front; gather loads
// run a 2-chunk-deep software pipeline ahead of the WMMAs.

typedef __attribute__((ext_vector_type(2))) float v2f;
typedef __attribute__((ext_vector_type(4))) float v4f;
typedef __attribute__((ext_vector_type(8))) float v8f;

#define BATCH 4
#define NROWS 50000
#define KDIM  32
#define DDIM  64
#define WPB   8                        // waves per block
#define BLKS_PER_BATCH (NROWS / WPB)   // 6250, exact

__device__ __forceinline__ int bperm_i(int src_lane, int v) {
  return __builtin_amdgcn_ds_bpermute(src_lane << 2, v);
}
__device__ __forceinline__ float bperm_f(int src_lane, float v) {
  return __int_as_float(__builtin_amdgcn_ds_bpermute(src_lane << 2, __float_as_int(v)));
}

__global__ __launch_bounds__(256)
void SparseMM_ell_wmma_kernel(const float* __restrict__ A,
                              const float* __restrict__ B,
                              const int*   __restrict__ IDX,
                              float*       __restrict__ OUT)
{
  const int lane = threadIdx.x & 31;
  const int wave = threadIdx.x >> 5;
  const int row  = blockIdx.x * WPB + wave;        // 0 .. BATCH*NROWS-1 (exact)
  const int bat  = blockIdx.x / BLKS_PER_BATCH;    // uniform (blocks never straddle)

  const float* Ab = A   + (size_t)row * KDIM;
  const int*   Ib = IDX + (size_t)row * KDIM;
  const char*  Bc = (const char*)(B + (size_t)bat * NROWS * DDIM);  // SGPR base

  // Lane k holds A[row][k] / idx[row][k] (K == 32 == wave width, coalesced).
  const float a_l = Ab[lane];
  const int   i_l = Ib[lane];

  const int      hs   = (lane >= 16) ? 2 : 0;      // upper half-wave = K rows k0+2/k0+3
  const bool     head = (lane & 15) == 0;          // lanes 0/16 carry A-matrix row M=0
  const unsigned colByte = (unsigned)((lane & 15) << 4);  // 16*j bytes within a row

  // ---- broadcast phase: all fragments + 32-bit byte voffsets, all 8 chunks ----
  v2f      afr[8];                                 // A fragments (row M=0 only)
  unsigned vbe[8], vbo[8];                         // byte offsets: idx*256 + 16*j
#pragma unroll
  for (int c = 0; c < 8; ++c) {
    const int s = 4 * c + hs;
    const float ae = bperm_f(s,     a_l);
    const float ao = bperm_f(s + 1, a_l);
    const int   ie = bperm_i(s,     i_l);
    const int   io = bperm_i(s + 1, i_l);
    afr[c].x = head ? ae : 0.0f;                   // K = k0   (lo half) / k0+2 (hi half)
    afr[c].y = head ? ao : 0.0f;                   // K = k0+1 / k0+3
    vbe[c] = ((unsigned)ie << 8) | colByte;        // one v_lshl_or_b32 each
    vbo[c] = ((unsigned)io << 8) | colByte;
  }

  v8f acc[4] = {};                                 // tile t = columns d % 4 == t

  // ---- gather + WMMA, 2-chunk-deep load pipeline (saddr + u32 voffset) ----
  v4f qe0 = *(const v4f*)(Bc + vbe[0]);
  v4f qo0 = *(const v4f*)(Bc + vbo[0]);
  v4f qe1 = *(const v4f*)(Bc + vbe[1]);
  v4f qo1 = *(const v4f*)(Bc + vbo[1]);

#pragma unroll
  for (int c = 0; c < 8; ++c) {
    const v4f qec = qe0, qoc = qo0;
    qe0 = qe1; qo0 = qo1;
    if (c < 6) {
      qe1 = *(const v4f*)(Bc + vbe[c + 2]);
      qo1 = *(const v4f*)(Bc + vbo[c + 2]);
    }
#pragma unroll
    for (int t = 0; t < 4; ++t) {
      v2f bf;
      bf.x = qec[t];                               // B[row_even][4j+t] -> B frag VGPR0
      bf.y = qoc[t];                               // B[row_odd ][4j+t] -> B frag VGPR1
      acc[t] = __builtin_amdgcn_wmma_f32_16x16x4_f32(
          /*neg_a=*/false, afr[c], /*neg_b=*/false, bf,
          /*c_mod=*/(short)0, acc[t], /*reuse_a=*/false, /*reuse_b=*/false);
    }
  }

  // D row M=0 lives in accumulator VGPR0, lanes 0-15 (N-col = lane = j).
  // Lane j holds out[4j + t] for t=0..3 -> contiguous float4 per lane.
  float* Ob = OUT + (size_t)row * DDIM;
  if (lane < 16) {
    v4f o;
    o.x = acc[0][0]; o.y = acc[1][0]; o.z = acc[2][0]; o.w = acc[3][0];
    __builtin_nontemporal_store(o, (v4f*)(Ob + (lane << 2)));
  }
}

extern "C" void kernel_launch(void* const* d_in, const int* in_sizes, int n_in,
                              void* d_out, int out_size, void* d_ws, size_t ws_size,
                              hipStream_t stream) {
  const float* A   = (const float*)d_in[0];
  const float* B   = (const float*)d_in[1];
  const int*   IDX = (const int*)d_in[2];
  float*       OUT = (float*)d_out;

  const int blocks = BATCH * BLKS_PER_BATCH;       // 25000, full waves only
  SparseMM_ell_wmma_kernel<<<blocks, 32 * WPB, 0, stream>>>(A, B, IDX, OUT);
}